// GPRGNN_34162169872617
// MI455X (gfx1250) — compile-verified
//
#include <hip/hip_runtime.h>
#include <math.h>

#define N_NODES 1216
#define GROUPS  19          // 1216 = 19 * 64
#define TPB     256
#define ROWS_PER_BLOCK 128  // 8 waves * 16 rows

typedef __attribute__((ext_vector_type(2))) float v2f;
typedef __attribute__((ext_vector_type(8))) float v8f;

// ---------------------------------------------------------------------------
// Kernel 1: one wave per node. Gather-count degree (deterministic, no float
// atomics), dinv = rsqrt(deg) (deg >= 1 due to self loop), init propagation
// state: cur = (w2, b2), u = temp0*w2, v = temp0*b2.
// ---------------------------------------------------------------------------
__global__ __launch_bounds__(TPB) void gpr_setup(
    const int* __restrict__ eidx, int E,
    const float* __restrict__ w2, const float* __restrict__ b2,
    const float* __restrict__ temp,
    float* __restrict__ dinv,
    float* __restrict__ curw, float* __restrict__ curb,
    float* __restrict__ u, float* __restrict__ v) {
  int tid  = threadIdx.x;
  int lane = tid & 31;
  int c = blockIdx.x * (TPB >> 5) + (tid >> 5);   // node handled by this wave
  if (c >= N_NODES) return;
  const int* col = eidx + E;
  float deg = 0.0f;
  for (int e = lane; e < E; e += 32) deg += (col[e] == c) ? 1.0f : 0.0f;
  #pragma unroll
  for (int off = 16; off > 0; off >>= 1) deg += __shfl_xor(deg, off, 32);
  if (lane == 0) {
    deg += 1.0f;                                   // self loop
    dinv[c] = rsqrtf(deg);
    float t0 = temp[0];
    float wv = w2[c], bv = b2[c];
    curw[c] = wv;       curb[c] = bv;
    u[c]    = t0 * wv;  v[c]    = t0 * bv;
  }
}

// ---------------------------------------------------------------------------
// Kernel 2: one propagation hop. One wave per destination node c:
//   new[c] = dinv[c]^2 * cur[c] + sum_{e: col_e==c} dinv[row_e]*dinv[c]*cur[row_e]
// for both the w- and b- vectors; then u += temp[step]*new_w, v += temp[step]*new_b.
// Fixed shuffle reduction order => bitwise deterministic across replays.
// ---------------------------------------------------------------------------
__global__ __launch_bounds__(TPB) void gpr_prop(
    const int* __restrict__ eidx, int E,
    const float* __restrict__ dinv,
    const float* __restrict__ curw, const float* __restrict__ curb,
    float* __restrict__ neww, float* __restrict__ newb,
    float* __restrict__ u, float* __restrict__ v,
    const float* __restrict__ temp, int step) {
  int tid  = threadIdx.x;
  int lane = tid & 31;
  int c = blockIdx.x * (TPB >> 5) + (tid >> 5);
  if (c >= N_NODES) return;
  const int* rowi = eidx;
  const int* coli = eidx + E;
  float dc = dinv[c];
  float aw = 0.0f, ab = 0.0f;
  for (int e = lane; e < E; e += 32) {
    if (coli[e] == c) {
      int r = rowi[e];
      float wgt = dinv[r] * dc;
      aw = fmaf(wgt, curw[r], aw);
      ab = fmaf(wgt, curb[r], ab);
    }
  }
  #pragma unroll
  for (int off = 16; off > 0; off >>= 1) {
    aw += __shfl_xor(aw, off, 32);
    ab += __shfl_xor(ab, off, 32);
  }
  if (lane == 0) {
    aw = fmaf(dc * dc, curw[c], aw);               // self loop
    ab = fmaf(dc * dc, curb[c], ab);
    neww[c] = aw; newb[c] = ab;
    float t = temp[step];
    u[c] = fmaf(t, aw, u[c]);
    v[c] = fmaf(t, ab, v[c]);
  }
}

// ---------------------------------------------------------------------------
// Kernel 3 (the bandwidth kernel, ~40 MB of traffic):
//  Phase A: h[b] = relu(x[b,:] . w1 + b1) via V_WMMA_F32_16X16X4_F32.
//    Each wave owns 16 batch rows. A tile = x[16 rows x 4 K] (exact ISA layout:
//    lane L<16 -> row L, K={0,1}; lane L+16 -> row L, K={2,3}). B tile = the
//    same 4 w1 values broadcast across all 16 columns using the SAME lane->K
//    map, so every column of D is the exact f32 dot product.
//  Phase B: out[b,n] = log_softmax_64( u[n]*h[b] + v[n] ), streamed out.
// ---------------------------------------------------------------------------
__global__ __launch_bounds__(TPB) void gpr_main(
    const float* __restrict__ x,    // (B, 1216) row-major
    const float* __restrict__ w1,   // 1216
    const float* __restrict__ b1p,  // scalar
    const float* __restrict__ u,
    const float* __restrict__ v,
    float* __restrict__ out) {
  __shared__ float u_s[N_NODES];
  __shared__ float v_s[N_NODES];
  __shared__ float w1_s[N_NODES];
  __shared__ float h_s[ROWS_PER_BLOCK];

  int tid = threadIdx.x;
  for (int i = tid; i < N_NODES; i += TPB) {
    u_s[i]  = u[i];
    v_s[i]  = v[i];
    w1_s[i] = w1[i];
  }
  __syncthreads();

  // ---- Phase A: WMMA GEMV ----
  int wave  = tid >> 5;
  int lane  = tid & 31;
  int mrow  = lane & 15;
  int khalf = (lane >> 4) * 2;      // lane-half -> K offset {0,2}, VGPR -> +0/+1
  long row  = (long)blockIdx.x * ROWS_PER_BLOCK + wave * 16 + mrow;
  const float* xr = x + row * N_NODES;

  v8f acc = {};
  for (int kb = 0; kb < N_NODES; kb += 4) {
    v2f a = *(const v2f*)(xr + kb + khalf);          // global_load_b64
    v2f b = *(const v2f*)(&w1_s[kb + khalf]);        // ds_load_b64 (broadcast)
    acc = __builtin_amdgcn_wmma_f32_16x16x4_f32(
        /*neg_a=*/false, a, /*neg_b=*/false, b,
        /*c_mod=*/(short)0, acc, /*reuse_a=*/false, /*reuse_b=*/false);
  }
  float b1 = b1p[0];
  if ((lane & 15) == 0) {           // lane 0 holds rows 0..7, lane 16 rows 8..15
    int mb = (lane >> 4) * 8;
    #pragma unroll
    for (int r = 0; r < 8; ++r) {
      float hv = acc[r] + b1;
      h_s[wave * 16 + mb + r] = hv > 0.0f ? hv : 0.0f;
    }
  }
  __syncthreads();

  // ---- Phase B: rank-1 expansion + group-64 log_softmax, streaming store ----
  long rowbase = (long)blockIdx.x * ROWS_PER_BLOCK;
  for (int p = tid; p < ROWS_PER_BLOCK * GROUPS; p += TPB) {
    int rl = p / GROUPS;
    int g  = p - rl * GROUPS;
    float h = h_s[rl];
    const float* ug = &u_s[g * 64];
    const float* vg = &v_s[g * 64];
    float mx = -INFINITY;
    #pragma unroll 4
    for (int j = 0; j < 64; ++j) mx = fmaxf(mx, fmaf(ug[j], h, vg[j]));
    float se = 0.0f;
    #pragma unroll 4
    for (int j = 0; j < 64; ++j) se += __expf(fmaf(ug[j], h, vg[j]) - mx);
    float lse = mx + __logf(se);
    float* o = out + (rowbase + rl) * N_NODES + g * 64;
    #pragma unroll 4
    for (int j = 0; j < 64; ++j) o[j] = fmaf(ug[j], h, vg[j]) - lse;
  }
}

// ---------------------------------------------------------------------------
extern "C" void kernel_launch(void* const* d_in, const int* in_sizes, int n_in,
                              void* d_out, int out_size, void* d_ws, size_t ws_size,
                              hipStream_t stream) {
  const float* x    = (const float*)d_in[0];
  const int*   eidx = (const int*)  d_in[1];   // (2, E)
  const float* w1   = (const float*)d_in[2];   // (1, 1216)
  const float* b1   = (const float*)d_in[3];   // scalar
  const float* w2   = (const float*)d_in[4];   // (1216, 1)
  const float* b2   = (const float*)d_in[5];   // (1216,)
  const float* temp = (const float*)d_in[6];   // (4,)
  float* out = (float*)d_out;

  int B = in_sizes[0] / N_NODES;               // 4096
  int E = in_sizes[1] / 2;                     // 38912

  float* ws   = (float*)d_ws;                  // 7*N floats of scratch
  float* dinv = ws;
  float* wA   = ws + 1 * N_NODES;
  float* wB   = ws + 2 * N_NODES;
  float* bA   = ws + 3 * N_NODES;
  float* bB   = ws + 4 * N_NODES;
  float* u    = ws + 5 * N_NODES;
  float* v    = ws + 6 * N_NODES;

  // one wave per node -> N_NODES*32 threads
  int nodeBlocks = (N_NODES * 32 + TPB - 1) / TPB;   // 152
  gpr_setup<<<nodeBlocks, TPB, 0, stream>>>(eidx, E, w2, b2, temp,
                                            dinv, wA, bA, u, v);
  gpr_prop<<<nodeBlocks, TPB, 0, stream>>>(eidx, E, dinv, wA, bA, wB, bB, u, v, temp, 1);
  gpr_prop<<<nodeBlocks, TPB, 0, stream>>>(eidx, E, dinv, wB, bB, wA, bA, u, v, temp, 2);
  gpr_prop<<<nodeBlocks, TPB, 0, stream>>>(eidx, E, dinv, wA, bA, wB, bB, u, v, temp, 3);

  gpr_main<<<B / ROWS_PER_BLOCK, TPB, 0, stream>>>(x, w1, b1, u, v, out);
}